// SE_semantic_Fusion_1606317769259
// MI455X (gfx1250) — compile-verified
//
#include <hip/hip_runtime.h>
#include <math.h>

typedef __attribute__((ext_vector_type(2))) float v2f;
typedef __attribute__((ext_vector_type(8))) float v8f;

#define NROW   128          // batch B
#define NCOLL  200704       // C*H*W = 256*784
#define NKB    128          // split-K blocks per tensor
#define SLAB   1568         // NCOLL / NKB
#define KC     196          // K-chunk staged in LDS
#define NCHUNK 8            // SLAB / KC
#define GSZ    16384        // 128*128

// workspace layout (float offsets)
#define WS_PARTIAL 0u
#define WS_G       4194304u             // + 2*128*16384
#define WS_Y       (WS_G + 32768u)      // y_s, y_t  (each 128*256)
#define WS_Z       (WS_Y + 65536u)      // z = [zs|zt]  (128*512)
#define WS_ZADD    (WS_Z + 65536u)      // 128*256
#define WS_ZMUL    (WS_ZADD + 32768u)   // 128*256
#define WS_GATE    (WS_ZMUL + 32768u)   // gate_s, gate_t (each 128*256)

// ---------------------------------------------------------------------------
// K1: split-K gram partials via V_WMMA_F32_16X16X4_F32
// grid (NKB, 2), block 256 (8 waves). Each wave: 2x4 grid of 16x16 tiles.
// Manual double-buffered software pipeline: in-place accumulators (no
// modulo-expansion register rotation / v_mov churn).
// ---------------------------------------------------------------------------
struct Frag { v2f a0, a1, b0, b1, b2, b3; };

__device__ __forceinline__ Frag ld_frag(const float* lds, int aoff0, int aoff1,
                                        int boff0, int boff1, int boff2, int boff3,
                                        int k) {
    Frag f;
    f.a0 = *(const v2f*)&lds[aoff0 + k];
    f.a1 = *(const v2f*)&lds[aoff1 + k];
    f.b0 = *(const v2f*)&lds[boff0 + k];
    f.b1 = *(const v2f*)&lds[boff1 + k];
    f.b2 = *(const v2f*)&lds[boff2 + k];
    f.b3 = *(const v2f*)&lds[boff3 + k];
    return f;
}

__device__ __forceinline__ void do_mma(v8f acc[8], const Frag& f) {
    acc[0] = __builtin_amdgcn_wmma_f32_16x16x4_f32(false, f.a0, false, f.b0, (short)0, acc[0], false, false);
    acc[1] = __builtin_amdgcn_wmma_f32_16x16x4_f32(false, f.a0, false, f.b1, (short)0, acc[1], false, false);
    acc[2] = __builtin_amdgcn_wmma_f32_16x16x4_f32(false, f.a0, false, f.b2, (short)0, acc[2], false, false);
    acc[3] = __builtin_amdgcn_wmma_f32_16x16x4_f32(false, f.a0, false, f.b3, (short)0, acc[3], false, false);
    acc[4] = __builtin_amdgcn_wmma_f32_16x16x4_f32(false, f.a1, false, f.b0, (short)0, acc[4], false, false);
    acc[5] = __builtin_amdgcn_wmma_f32_16x16x4_f32(false, f.a1, false, f.b1, (short)0, acc[5], false, false);
    acc[6] = __builtin_amdgcn_wmma_f32_16x16x4_f32(false, f.a1, false, f.b2, (short)0, acc[6], false, false);
    acc[7] = __builtin_amdgcn_wmma_f32_16x16x4_f32(false, f.a1, false, f.b3, (short)0, acc[7], false, false);
}

__global__ __launch_bounds__(256) void k_gram(const float* __restrict__ stu,
                                              const float* __restrict__ tea,
                                              float* __restrict__ partial) {
    __shared__ float lds[NROW * KC];    // 100,352 bytes
    const int tid  = threadIdx.x;
    const int kb   = blockIdx.x;
    const int side = blockIdx.y;
    const float* __restrict__ src = side ? tea : stu;

    const int lane = tid & 31;
    const int wv   = tid >> 5;
    const int I2   = wv & 3;            // tile-row pair index (rows 2*I2, 2*I2+1)
    const int J4   = wv >> 2;           // tile-col quad index (cols 4*J4..4*J4+3)
    const int m    = lane & 15;
    const int hi   = lane >> 4;

    v8f acc[8] = {};                    // [a(2)][b(4)]

    const int aoff0 = (32 * I2 + m) * KC;
    const int aoff1 = aoff0 + 16 * KC;
    const int boff0 = (64 * J4 + m) * KC;
    const int boff1 = boff0 + 16 * KC;
    const int boff2 = boff0 + 32 * KC;
    const int boff3 = boff0 + 48 * KC;
    const int kbase = 2 * hi;

    for (int ch = 0; ch < NCHUNK; ++ch) {
        const size_t k0 = (size_t)kb * SLAB + (size_t)ch * KC;
        __syncthreads();
        // stage 128 x 196 floats, coalesced float2 loads
        for (int i = tid; i < NROW * (KC / 2); i += 256) {
            const int r  = i / (KC / 2);
            const int c2 = i - r * (KC / 2);
            const float2 v = *(const float2*)(src + (size_t)r * NCOLL + k0 + 2 * c2);
            *(float2*)&lds[r * KC + 2 * c2] = v;
        }
        __syncthreads();

        // 49 K-steps of 4: manual double-buffered pipeline (24 pairs + tail)
        Frag f0 = ld_frag(lds, aoff0, aoff1, boff0, boff1, boff2, boff3, kbase);
        #pragma unroll 1
        for (int it = 0; it < 24; ++it) {
            Frag f1 = ld_frag(lds, aoff0, aoff1, boff0, boff1, boff2, boff3,
                              kbase + 4 * (2 * it + 1));
            do_mma(acc, f0);
            f0 = ld_frag(lds, aoff0, aoff1, boff0, boff1, boff2, boff3,
                         kbase + 4 * (2 * it + 2));
            do_mma(acc, f1);
        }
        do_mma(acc, f0);                // K-step 48
    }

    float* __restrict__ p = partial + (size_t)(side * NKB + kb) * GSZ;
    #pragma unroll
    for (int ta = 0; ta < 2; ++ta) {
        #pragma unroll
        for (int tb = 0; tb < 4; ++tb) {
            const int rowb = 32 * I2 + 16 * ta;
            const int colb = 64 * J4 + 16 * tb;
            const v8f c = acc[ta * 4 + tb];
            #pragma unroll
            for (int v = 0; v < 8; ++v)
                p[(rowb + v + 8 * hi) * 128 + colb + m] = c[v];
        }
    }
}

// ---------------------------------------------------------------------------
// K2: reduce partials -> G_s, G_t.  grid 128, block 256
// ---------------------------------------------------------------------------
__global__ __launch_bounds__(256) void k_reduce(const float* __restrict__ partial,
                                                float* __restrict__ G) {
    const int idx  = blockIdx.x * 256 + threadIdx.x;   // [0, 32768)
    const int side = idx >> 14;
    const int e    = idx & (GSZ - 1);
    float s = 0.f;
    const float* __restrict__ p = partial + (size_t)side * NKB * GSZ + e;
    for (int kb = 0; kb < NKB; ++kb) s += p[(size_t)kb * GSZ];
    G[idx] = s;
}

// ---------------------------------------------------------------------------
// K3: channel means. grid (32,128,2), block 256 (8 waves); one wave per channel
// ---------------------------------------------------------------------------
__global__ __launch_bounds__(256) void k_means(const float* __restrict__ stu,
                                               const float* __restrict__ tea,
                                               float* __restrict__ y) {
    const int lane = threadIdx.x & 31;
    const int wv   = threadIdx.x >> 5;
    const int c    = blockIdx.x * 8 + wv;
    const int b    = blockIdx.y;
    const int side = blockIdx.z;
    const float* __restrict__ src = side ? tea : stu;
    const size_t base = (size_t)(b * 256 + c) * 784;
    float s = 0.f;
    for (int i = lane; i < 784; i += 32) s += src[base + i];
    #pragma unroll
    for (int off = 16; off > 0; off >>= 1) s += __shfl_xor(s, off, 32);
    if (lane == 0) y[side * 32768 + b * 256 + c] = s * (1.f / 784.f);
}

// ---------------------------------------------------------------------------
// K4: row-normalize G, z = G~ @ y, z_add, z_mul. grid 128 (row b), block 256
// ---------------------------------------------------------------------------
__global__ __launch_bounds__(256) void k_zvec(const float* __restrict__ G,
                                              const float* __restrict__ y,
                                              float* __restrict__ z,
                                              float* __restrict__ z_add,
                                              float* __restrict__ z_mul) {
    __shared__ float sh[256];
    const int t = threadIdx.x;
    const int b = blockIdx.x;
    const float* __restrict__ Gs = G + b * 128;
    const float* __restrict__ Gt = G + GSZ + b * 128;
    {
        float v = (t < 128) ? Gs[t] : Gt[t - 128];
        sh[t] = v * v;
    }
    __syncthreads();
    for (int s = 64; s > 0; s >>= 1) {
        if (t < s) sh[t] += sh[t + s];
        else if (t >= 128 && t < 128 + s) sh[t] += sh[t + s];
        __syncthreads();
    }
    const float inv_s = 1.f / fmaxf(sqrtf(sh[0]),   1e-12f);
    const float inv_t = 1.f / fmaxf(sqrtf(sh[128]), 1e-12f);

    const float* __restrict__ ys = y;
    const float* __restrict__ yt = y + 32768;
    float zs = 0.f, zt = 0.f;
    for (int k = 0; k < 128; ++k) {
        zs += Gs[k] * ys[k * 256 + t];
        zt += Gt[k] * yt[k * 256 + t];
    }
    zs *= inv_s;
    zt *= inv_t;
    z[b * 512 + t]        = zs;
    z[b * 512 + 256 + t]  = zt;
    z_add[b * 256 + t]    = zs + zt;
    z_mul[b * 256 + t]    = zs * zt;
}

// ---------------------------------------------------------------------------
// K5: six tiny MLPs -> gates. grid (128, 2), block 256 (thread = out channel)
// ---------------------------------------------------------------------------
struct MlpAll { const float* p[24]; };   // 6 mlps x (w1,b1,w2,b2)

__global__ __launch_bounds__(256) void k_mlp(MlpAll wts,
                                             const float* __restrict__ z,
                                             const float* __restrict__ z_add,
                                             const float* __restrict__ z_mul,
                                             float* __restrict__ gates) {
    __shared__ float s_in[512];
    __shared__ float s_h[128];
    const int t    = threadIdx.x;
    const int b    = blockIdx.x;
    const int side = blockIdx.y;
    const int base = side * 12;
    float gacc = 0.f;

    for (int mi = 0; mi < 3; ++mi) {
        const int din = (mi == 0) ? 512 : 256;
        const int dh  = (mi == 0) ? 128 : 64;
        const float* __restrict__ in =
            (mi == 0) ? (z + b * 512) : ((mi == 1) ? (z_add + b * 256) : (z_mul + b * 256));
        const float* __restrict__ w1 = wts.p[base + mi * 4 + 0];
        const float* __restrict__ b1 = wts.p[base + mi * 4 + 1];
        const float* __restrict__ w2 = wts.p[base + mi * 4 + 2];
        const float* __restrict__ b2 = wts.p[base + mi * 4 + 3];

        for (int i = t; i < din; i += 256) s_in[i] = in[i];
        __syncthreads();
        if (t < dh) {
            float a = b1[t];
            const float* __restrict__ wr = w1 + t * din;
            for (int d = 0; d < din; ++d) a += s_in[d] * wr[d];
            s_h[t] = a > 0.f ? a : 0.f;
        }
        __syncthreads();
        {
            float o = b2[t];
            const float* __restrict__ wr = w2 + t * dh;
            for (int j = 0; j < dh; ++j) o += s_h[j] * wr[j];
            gacc += 1.f / (1.f + __expf(-o));
        }
        __syncthreads();
    }
    gates[side * 32768 + b * 256 + t] = gacc;
}

// ---------------------------------------------------------------------------
// K6: apply gates. float4 streaming. grid 50176, block 256
// ---------------------------------------------------------------------------
__global__ __launch_bounds__(256) void k_apply(const float* __restrict__ stu,
                                               const float* __restrict__ tea,
                                               const float* __restrict__ gates,
                                               float* __restrict__ out) {
    const int idx  = blockIdx.x * 256 + threadIdx.x;   // float4 index, exact
    const int chan = idx / 196;                        // (side,b,c) flat
    const int rem  = idx - chan * 196;
    const float g  = gates[chan];
    const int side = chan >> 15;
    const int wi   = chan & 32767;
    const float* __restrict__ src = side ? tea : stu;
    const size_t off = (size_t)wi * 784 + (size_t)rem * 4;
    float4 v = *(const float4*)(src + off);
    v.x *= g; v.y *= g; v.z *= g; v.w *= g;
    *(float4*)(out + (size_t)idx * 4) = v;
}

// ---------------------------------------------------------------------------
extern "C" void kernel_launch(void* const* d_in, const int* in_sizes, int n_in,
                              void* d_out, int out_size, void* d_ws, size_t ws_size,
                              hipStream_t stream) {
    const float* stu = (const float*)d_in[0];
    const float* tea = (const float*)d_in[1];
    float* ws  = (float*)d_ws;
    float* out = (float*)d_out;

    float* partial = ws + WS_PARTIAL;
    float* G       = ws + WS_G;
    float* y       = ws + WS_Y;
    float* z       = ws + WS_Z;
    float* z_add   = ws + WS_ZADD;
    float* z_mul   = ws + WS_ZMUL;
    float* gates   = ws + WS_GATE;

    MlpAll wts;
    for (int i = 0; i < 24; ++i) wts.p[i] = (const float*)d_in[2 + i];

    k_gram  <<<dim3(NKB, 2),       256, 0, stream>>>(stu, tea, partial);
    k_reduce<<<dim3(128),          256, 0, stream>>>(partial, G);
    k_means <<<dim3(32, 128, 2),   256, 0, stream>>>(stu, tea, y);
    k_zvec  <<<dim3(128),          256, 0, stream>>>(G, y, z, z_add, z_mul);
    k_mlp   <<<dim3(128, 2),       256, 0, stream>>>(wts, z, z_add, z_mul, gates);
    k_apply <<<dim3(50176),        256, 0, stream>>>(stu, tea, gates, out);
}